// FieldTheoryAttention_21509196218750
// MI455X (gfx1250) — compile-verified
//
#include <hip/hip_runtime.h>
#include <hip/hip_bf16.h>

typedef float v2f __attribute__((ext_vector_type(2)));
typedef float v4f __attribute__((ext_vector_type(4)));
typedef float v8f __attribute__((ext_vector_type(8)));

#define Dn 384
#define NELT (Dn * Dn) /* 147456 */

// ---------------- block reduction helpers (power-of-two blockDim) ----------
__device__ __forceinline__ float bsum(float* sm, int tid, int n, float v) {
  sm[tid] = v; __syncthreads();
  for (int s = n >> 1; s > 0; s >>= 1) { if (tid < s) sm[tid] += sm[tid + s]; __syncthreads(); }
  float r = sm[0]; __syncthreads();
  return r;
}
__device__ __forceinline__ float bmax(float* sm, int tid, int n, float v) {
  sm[tid] = v; __syncthreads();
  for (int s = n >> 1; s > 0; s >>= 1) { if (tid < s) sm[tid] = fmaxf(sm[tid], sm[tid + s]); __syncthreads(); }
  float r = sm[0]; __syncthreads();
  return r;
}

// ---------------- scratch accumulator zeroing ------------------------------
__global__ void zero_sc(float* SC) { if (threadIdx.x < 16) SC[threadIdx.x] = 0.0f; }

// ---------------- prep: gradients, commutator-mean, row sums ---------------
// One block per row i (128 threads, 3 cols/thread).
// SC: [0]=cm_total [1]=init_e
__global__ void prep(const float* __restrict__ q, const float* __restrict__ k,
                     float* __restrict__ GA, float* __restrict__ GB, float* __restrict__ CM,
                     float* __restrict__ GARS, float* __restrict__ GBRS, float* SC) {
  __shared__ float sm[128];
  const int i = blockIdx.x, tid = threadIdx.x;
  float qv[3], kv[3], ga[3], gb[3];
  for (int c = 0; c < 3; ++c) {
    const int d = tid + c * 128;
    qv[c] = q[i * Dn + d];
    kv[c] = k[i * Dn + d];
    if (i == 0)            { ga[c] = q[Dn + d] - q[d];                           gb[c] = k[Dn + d] - k[d]; }
    else if (i == Dn - 1)  { ga[c] = q[(Dn-1)*Dn + d] - q[(Dn-2)*Dn + d];        gb[c] = k[(Dn-1)*Dn + d] - k[(Dn-2)*Dn + d]; }
    else                   { ga[c] = 0.5f * (q[(i+1)*Dn + d] - q[(i-1)*Dn + d]); gb[c] = 0.5f * (k[(i+1)*Dn + d] - k[(i-1)*Dn + d]); }
    GA[i * Dn + d] = ga[c];
    GB[i * Dn + d] = gb[c];
  }
  const float qsum = bsum(sm, tid, 128, qv[0] + qv[1] + qv[2]);
  const float ksum = bsum(sm, tid, 128, kv[0] + kv[1] + kv[2]);
  const float qbar = qsum * (1.0f / Dn), kbar = ksum * (1.0f / Dn);
  float csum_l = 0.0f, qsq_l = 0.0f;
  for (int c = 0; c < 3; ++c) {
    const int d = tid + c * 128;
    const float cm = qv[c] * kbar - kv[c] * qbar;
    CM[i * Dn + d] = cm;
    csum_l += cm;
    qsq_l += qv[c] * qv[c];
  }
  const float csum = bsum(sm, tid, 128, csum_l);
  const float qsq  = bsum(sm, tid, 128, qsq_l);
  const float gas  = bsum(sm, tid, 128, ga[0] + ga[1] + ga[2]);
  const float gbs  = bsum(sm, tid, 128, gb[0] + gb[1] + gb[2]);
  if (tid == 0) {
    GARS[i] = gas;
    GBRS[i] = gbs;
    atomicAdd(SC + 0, csum);
    atomicAdd(SC + 1, qsq);
  }
}

// ---------------- generic fp32 WMMA GEMM (one wave = one 16x16 tile) -------
// Out[m,n] = op( scale * sum_k A[m,k]*B[k,n]  (+ Cadd[m,n]) (+ bias[n]) )
// BT=1 => B accessed as B[n*ldb + k] (i.e. logits = covq @ covk^T).
// All mode flags are template params -> branch-free inner loop:
//   global_load_b64 (A) + loads (B) + v_wmma_f32_16x16x4_f32 + pointer bumps.
template <int BT, int HASADD, int HASBIAS, int RELU>
__global__ __launch_bounds__(32) void gemm_wmma(
    const float* __restrict__ A, int lda,
    const float* __restrict__ B, int ldb,
    const float* __restrict__ Cadd, int ldadd,
    const float* __restrict__ bias,
    float scale,
    float* __restrict__ Out, int ldo, int K) {
  const int tileM = blockIdx.y * 16;
  const int tileN = blockIdx.x * 16;
  const int lane = threadIdx.x;        // 0..31, full wave (EXEC all ones)
  const int half = lane >> 4;          // 0 or 1
  const int l15  = lane & 15;

  // per-lane base pointers, bumped incrementally (no per-k address math)
  const float* ap = A + (size_t)(tileM + l15) * lda + half * 2;
  const float* bp = BT ? (B + (size_t)(tileN + l15) * ldb + half * 2)
                       : (B + (size_t)(half * 2) * ldb + (tileN + l15));
  const size_t bstep = BT ? 4 : (size_t)4 * ldb;

  v8f acc = {};
  #pragma unroll 4
  for (int k = 0; k < K; k += 4) {
    const v2f a = *(const v2f*)ap;           // 8B aligned: lda even, k%4==0
    v2f b;
    if (BT) {
      b = *(const v2f*)bp;
    } else {
      b.x = bp[0];
      b.y = bp[ldb];
    }
    acc = __builtin_amdgcn_wmma_f32_16x16x4_f32(false, a, false, b, (short)0, acc, false, false);
    ap += 4;
    bp += bstep;
  }

  const int n = tileN + l15;
  #pragma unroll
  for (int r = 0; r < 8; ++r) {
    const int m = tileM + r + half * 8;
    float vz = acc[r] * scale;
    if (HASADD)  vz += Cadd[(size_t)m * ldadd + n];
    if (HASBIAS) vz += bias[n];
    if (RELU)    vz = fmaxf(vz, 0.0f);
    Out[(size_t)m * ldo + n] = vz;
  }
}

// ---------------- row softmax (in place), records post-normalization sum ---
__global__ void softmax_rows(float* __restrict__ LG, float* __restrict__ RS) {
  __shared__ float sm[128];
  const int r = blockIdx.x, tid = threadIdx.x;
  float x0 = LG[r * Dn + tid], x1 = LG[r * Dn + tid + 128], x2 = LG[r * Dn + tid + 256];
  const float m = bmax(sm, tid, 128, fmaxf(x0, fmaxf(x1, x2)));
  x0 = expf(x0 - m); x1 = expf(x1 - m); x2 = expf(x2 - m);
  const float s = bsum(sm, tid, 128, x0 + x1 + x2);
  const float inv = 1.0f / s;
  x0 *= inv; x1 *= inv; x2 *= inv;
  LG[r * Dn + tid] = x0; LG[r * Dn + tid + 128] = x1; LG[r * Dn + tid + 256] = x2;
  const float rs = bsum(sm, tid, 128, x0 + x1 + x2);
  if (tid == 0) RS[r] = rs;
}

// ---------------- interaction density: SC[4] += dot(h[i,:], w2) ------------
__global__ void density(const float* __restrict__ H, const float* __restrict__ w2, float* SC) {
  __shared__ float sm[128];
  const int i = blockIdx.x, tid = threadIdx.x;
  float l = H[i * Dn + tid]       * w2[tid]
          + H[i * Dn + tid + 128] * w2[tid + 128]
          + H[i * Dn + tid + 256] * w2[tid + 256];
  const float s = bsum(sm, tid, 128, l);
  if (tid == 0) atomicAdd(SC + 4, s);
}

// ---------------- SC[3] += sum(E^2) ----------------------------------------
__global__ void esq(const float* __restrict__ E, float* SC) {
  __shared__ float sm[256];
  const int idx = blockIdx.x * 256 + threadIdx.x;
  const float v = E[idx];
  const float s = bsum(sm, threadIdx.x, 256, v * v);
  if (threadIdx.x == 0) atomicAdd(SC + 3, s);
}

// ---------------- field_strength writer (b128 NT stores + fused YM) --------
// F[s,i,j] = GA[s,i] - GB[s,j] + CM[i,j];  SC[2] += sum F^2
__global__ void fs_write(const float* __restrict__ GA, const float* __restrict__ GB,
                         const float* __restrict__ CM, float* __restrict__ out, float* SC) {
  __shared__ float sm[256];
  const int s = blockIdx.y;
  const int idx4 = blockIdx.x * 256 + threadIdx.x;   // 0..36863
  const int i = idx4 / 96;
  const int j = (idx4 % 96) * 4;
  const float gav = GA[s * Dn + i];
  const v4f cmv = *(const v4f*)(CM + i * Dn + j);
  const v4f gbv = *(const v4f*)(GB + s * Dn + j);
  v4f f;
  f.x = gav - gbv.x + cmv.x;
  f.y = gav - gbv.y + cmv.y;
  f.z = gav - gbv.z + cmv.z;
  f.w = gav - gbv.w + cmv.w;
  __builtin_nontemporal_store(f, (v4f*)(out + (size_t)s * NELT + (size_t)i * Dn + j));
  const float sq = f.x * f.x + f.y * f.y + f.z * f.z + f.w * f.w;
  const float tot = bsum(sm, threadIdx.x, 256, sq);
  if (threadIdx.x == 0) atomicAdd(SC + 2, tot);
}

// ---------------- evolved_field writer: replicate E 384x with NT stores ----
__global__ void ev_write(const float* __restrict__ E, float* __restrict__ out) {
  const int s = blockIdx.y;
  const int idx4 = blockIdx.x * 256 + threadIdx.x;
  const v4f ev = *(const v4f*)(E + (size_t)idx4 * 4);
  __builtin_nontemporal_store(ev, (v4f*)(out + (size_t)s * NELT + (size_t)idx4 * 4));
}

// ---------------- finalize: all small outputs ------------------------------
__global__ void finalize(const float* __restrict__ RS, const float* __restrict__ GARS,
                         const float* __restrict__ GBRS, const float* SC,
                         const float* __restrict__ coup, const float* __restrict__ b2,
                         float* o_ie, float* o_ym, float* o_fc, float* o_gi,
                         float* o_cons, float* o_coh, float* o_topo) {
  __shared__ float sm[512];
  const int t = threadIdx.x;
  const float rsv = (t < Dn) ? RS[t] : 0.0f;
  const float tot = bsum(sm, t, 512, rsv);
  const float mean = tot * (1.0f / Dn);
  const float d = (t < Dn) ? (rsv - mean) : 0.0f;
  const float ssd = bsum(sm, t, 512, d * d);
  const float sga = bsum(sm, t, 512, (t < Dn) ? GARS[t] : 0.0f);
  const float sgb = bsum(sm, t, 512, (t < Dn) ? GBRS[t] : 0.0f);
  const float init_e = SC[1], fin_e = SC[3];
  if (t < Dn) {
    o_fc[t] = 0.0f;  // evolved_field is constant along seq axis -> second grad == 0
    o_cons[t] = fabsf(fin_e - init_e) / (init_e + 1e-8f);
  }
  if (t == 0) {
    o_ie[0] = SC[4] + (float)Dn * b2[0];
    o_ym[0] = coup[0] * SC[2];
    const float var = ((float)Dn * ssd) / ((float)(Dn * Dn) - 1.0f);
    float gi = 1.0f - sqrtf(var);
    gi = fminf(fmaxf(gi, 0.0f), 1.0f);
    o_gi[0] = gi;
    o_coh[0] = 1.0f;  // corrcoef of a single row is [[1]]
    o_topo[0] = (sga - sgb + SC[0]) * (1.0f / 6.283185307179586f);
  }
}

// ---------------------------------------------------------------------------
extern "C" void kernel_launch(void* const* d_in, const int* in_sizes, int n_in,
                              void* d_out, int out_size, void* d_ws, size_t ws_size,
                              hipStream_t stream) {
  const float* q    = (const float*)d_in[0];
  const float* k    = (const float*)d_in[1];
  const float* v    = (const float*)d_in[2];
  // d_in[3] = gauge_weights: provably cancelled by softmax shift-invariance.
  const float* Wc   = (const float*)d_in[4];
  const float* coup = (const float*)d_in[5];
  const float* w1   = (const float*)d_in[6];
  const float* b1   = (const float*)d_in[7];
  const float* w2   = (const float*)d_in[8];
  const float* b2   = (const float*)d_in[9];

  float* W    = (float*)d_ws;
  float* GA   = W;                   // [384,384] grad_seq(q)
  float* GB   = GA + NELT;           // [384,384] grad_seq(k)
  float* CM   = GB + NELT;           // [384,384] commutator mean
  float* COMB = CM + NELT;           // [384,768] = [cov_q | cov_k]
  float* LG   = COMB + 2 * NELT;     // [384,384] logits -> softmax in place
  float* EB   = LG + NELT;           // [384,384] E = A @ V
  float* HB   = EB + NELT;           // [384,384] relu MLP hidden
  float* RS   = HB + NELT;           // [384] attention row sums
  float* GARS = RS + Dn;             // [384]
  float* GBRS = GARS + Dn;           // [384]
  float* SC   = GBRS + Dn;           // [16] scalar accumulators

  float* out    = (float*)d_out;
  float* o_ev   = out;                              // [1,384,384,384]
  float* o_fs   = o_ev + (size_t)Dn * NELT;         // [1,384,384,384]
  float* o_ie   = o_fs + (size_t)Dn * NELT;         // [1,1]
  float* o_ym   = o_ie + 1;                         // [1]
  float* o_fc   = o_ym + 1;                         // [1,384]
  float* o_gi   = o_fc + Dn;                        // scalar
  float* o_cons = o_gi + 1;                         // [1,384]
  float* o_coh  = o_cons + Dn;                      // scalar
  float* o_topo = o_coh + 1;                        // [1]

  const float inv_sqrt_d = 0.051031036307982884f;   // 1/sqrt(384)

  zero_sc<<<1, 16, 0, stream>>>(SC);
  prep<<<Dn, 128, 0, stream>>>(q, k, GA, GB, CM, GARS, GBRS, SC);

  dim3 g24(24, 24);
  // cov_q = grad(q) + q @ Wc  -> COMB[:, 0:384]
  gemm_wmma<0, 1, 0, 0><<<g24, 32, 0, stream>>>(q, Dn, Wc, Dn, GA, Dn, nullptr, 1.0f,
                                                COMB, 2 * Dn, Dn);
  // cov_k = grad(k) + k @ Wc  -> COMB[:, 384:768]
  gemm_wmma<0, 1, 0, 0><<<g24, 32, 0, stream>>>(k, Dn, Wc, Dn, GB, Dn, nullptr, 1.0f,
                                                COMB + Dn, 2 * Dn, Dn);
  // logits = cov_q @ cov_k^T / sqrt(D)
  gemm_wmma<1, 0, 0, 0><<<g24, 32, 0, stream>>>(COMB, 2 * Dn, COMB + Dn, 2 * Dn, nullptr, 0,
                                                nullptr, inv_sqrt_d, LG, Dn, Dn);
  softmax_rows<<<Dn, 128, 0, stream>>>(LG, RS);
  // E = A @ V
  gemm_wmma<0, 0, 0, 0><<<g24, 32, 0, stream>>>(LG, Dn, v, Dn, nullptr, 0, nullptr, 1.0f,
                                                EB, Dn, Dn);
  // H = relu([cov_q|cov_k] @ W1 + b1)
  gemm_wmma<0, 0, 1, 1><<<g24, 32, 0, stream>>>(COMB, 2 * Dn, w1, Dn, nullptr, 0, b1, 1.0f,
                                                HB, Dn, 2 * Dn);
  density<<<Dn, 128, 0, stream>>>(HB, w2, SC);
  esq<<<NELT / 256, 256, 0, stream>>>(EB, SC);

  dim3 gw(144, Dn);  // 144*256 threads * 4 floats = one 384x384 slice
  fs_write<<<gw, 256, 0, stream>>>(GA, GB, CM, o_fs, SC);
  ev_write<<<gw, 256, 0, stream>>>(EB, o_ev);

  finalize<<<1, 512, 0, stream>>>(RS, GARS, GBRS, SC, coup, b2,
                                  o_ie, o_ym, o_fc, o_gi, o_cons, o_coh, o_topo);
}